// RNN_Decoder_15238543966797
// MI455X (gfx1250) — compile-verified
//
#include <hip/hip_runtime.h>
#include <math.h>

typedef float v2f __attribute__((ext_vector_type(2)));
typedef float v8f __attribute__((ext_vector_type(8)));
typedef unsigned int u32x4 __attribute__((ext_vector_type(4)));
typedef int i32x4 __attribute__((ext_vector_type(4)));
typedef int i32x8 __attribute__((ext_vector_type(8)));

// ---------------------------------------------------------------------------
// TDM: issue a 2D f32 tile load (tile_d0 x tile_d1 elements, row stride
// stride0 elements) from global memory into LDS. Descriptor per CDNA5 ISA
// 8.3/8.4. Issue and completion-wait are split so the DMA overlaps compute;
// the issuing wave drains TENSORcnt before the workgroup barrier.
// ---------------------------------------------------------------------------
#if __has_builtin(__builtin_amdgcn_tensor_load_to_lds)
#define HAVE_TDM 1
__device__ inline void tdm_issue_tile_f32(unsigned int lds_byte_off, unsigned long long gaddr,
                                          unsigned int tensor_d0, unsigned int tensor_d1,
                                          unsigned int tile_d0, unsigned int tile_d1,
                                          unsigned int stride0)
{
    u32x4 g0;
    g0[0] = 1u;                                              // count=1, user descriptor
    g0[1] = lds_byte_off;                                    // lds_addr (bytes)
    g0[2] = (unsigned int)(gaddr & 0xFFFFFFFFu);             // global_addr[31:0]
    g0[3] = (unsigned int)((gaddr >> 32) & 0x01FFFFFFu)      // global_addr[56:32]
          | (2u << 30);                                      // type = 2 ("image")
    i32x8 g1;
    g1[0] = (int)(2u << 16);                                 // data_size=2 -> 4 bytes
    g1[1] = (int)((tensor_d0 & 0xFFFFu) << 16);              // tensor_dim0[15:0]
    g1[2] = (int)(((tensor_d0 >> 16) & 0xFFFFu)
          | ((tensor_d1 & 0xFFFFu) << 16));                  // tensor_dim1[15:0]
    g1[3] = (int)(((tensor_d1 >> 16) & 0xFFFFu)
          | ((tile_d0 & 0xFFFFu) << 16));                    // tile_dim0
    g1[4] = (int)(tile_d1 & 0xFFFFu);                        // tile_dim1 (tile_dim2=0)
    g1[5] = (int)stride0;                                    // tensor_dim0_stride[31:0]
    g1[6] = 0;
    g1[7] = 0;
    i32x4 z4 = {0, 0, 0, 0};
#if defined(__clang_major__) && (__clang_major__ >= 23)
    i32x8 z8 = {0, 0, 0, 0, 0, 0, 0, 0};
    __builtin_amdgcn_tensor_load_to_lds(g0, g1, z4, z4, z8, 0);
#else
    __builtin_amdgcn_tensor_load_to_lds(g0, g1, z4, z4, 0);
#endif
}
__device__ inline void tdm_drain() { __builtin_amdgcn_s_wait_tensorcnt(0); }
#else
#define HAVE_TDM 0
#endif

// ---------------------------------------------------------------------------
// Generic M=64 WMMA-f32 GEMM:  C[b] = A[b] (64xK) * B[b] (KxN) [+ A2*B2] [+ bias]
// Element B[k][n] lives at B + k*bk + n*bn. One wave32 owns one 16x16 C tile.
// Software pipeline with a *peeled* final iteration: the loop body always
// preloads the next fragments (no guard branch, no rotation movs), so each
// WMMA only waits on loads issued one iteration earlier.
// ---------------------------------------------------------------------------
__global__ void __launch_bounds__(256)
wmma_gemm64(const float* __restrict__ A,  long lda, long aBatch,
            const float* __restrict__ B,  long bk, long bn, long bBatch,
            const float* __restrict__ A2, const float* __restrict__ B2,
            const float* __restrict__ bias,
            float* __restrict__ C, long ldc, long cBatch,
            int N, int K, int nbatch)
{
    const int ntiles        = N >> 4;
    const int wavesPerBatch = ntiles << 2;              // 4 M-tiles (M = 64)
    const int wave = (int)((blockIdx.x * blockDim.x + threadIdx.x) >> 5);
    const int lane = (int)(threadIdx.x & 31);
    const int batch = wave / wavesPerBatch;
    if (batch >= nbatch) return;                        // wave-uniform exit
    const int t    = wave - batch * wavesPerBatch;
    const int mt   = (t & 3) << 4;
    const int nt   = (t >> 2) << 4;
    const int half = lane >> 4;
    const int l16  = lane & 15;

    const size_t bstep = (size_t)4 * bk;
    const size_t pref  = (size_t)128 * bk;

    const float* ap = A + (size_t)batch * aBatch + (size_t)(mt + l16) * lda + 2 * half;
    const float* bp = B + (size_t)batch * bBatch + (size_t)(nt + l16) * bn
                        + (size_t)(2 * half) * bk;

    v8f acc = {};
    {
        v2f av, bv;
        av.x = ap[0];  av.y = ap[1];
        bv.x = bp[0];  bv.y = bp[bk];
        for (int k = 0; k < K - 4; k += 4) {
            ap += 4;  bp += bstep;
            if ((k & 63) == 0)
                __builtin_prefetch((const void*)(bp + pref), 0, 0);
            v2f an, bn;                                 // unconditional preload
            an.x = ap[0];  an.y = ap[1];
            bn.x = bp[0];  bn.y = bp[bk];
            acc = __builtin_amdgcn_wmma_f32_16x16x4_f32(false, av, false, bv,
                                                        (short)0, acc, false, false);
            av = an;  bv = bn;
        }
        acc = __builtin_amdgcn_wmma_f32_16x16x4_f32(false, av, false, bv,
                                                    (short)0, acc, false, false);
    }
    if (A2 != nullptr) {                                // fused second product
        const float* ap2 = A2 + (size_t)(mt + l16) * lda + 2 * half;
        const float* bp2 = B2 + (size_t)(nt + l16) * bn + (size_t)(2 * half) * bk;
        v2f av, bv;
        av.x = ap2[0];  av.y = ap2[1];
        bv.x = bp2[0];  bv.y = bp2[bk];
        for (int k = 0; k < K - 4; k += 4) {
            ap2 += 4;  bp2 += bstep;
            if ((k & 63) == 0)
                __builtin_prefetch((const void*)(bp2 + pref), 0, 0);
            v2f an, bn;
            an.x = ap2[0];  an.y = ap2[1];
            bn.x = bp2[0];  bn.y = bp2[bk];
            acc = __builtin_amdgcn_wmma_f32_16x16x4_f32(false, av, false, bv,
                                                        (short)0, acc, false, false);
            av = an;  bv = bn;
        }
        acc = __builtin_amdgcn_wmma_f32_16x16x4_f32(false, av, false, bv,
                                                    (short)0, acc, false, false);
    }

    const float bval = bias ? bias[nt + l16] : 0.0f;
    float* c = C + (size_t)batch * cBatch + (size_t)nt + l16;
    #pragma unroll
    for (int r = 0; r < 8; ++r) {
        const int m = mt + r + 8 * half;
        c[(size_t)m * ldc] = acc[r] + bval;
    }
}

// ---------------------------------------------------------------------------
// Logits GEMM: C(64x32000) = A(64x1024) @ W2(1024x32000) + b2.
// Each wave owns a 64x16 strip (4 accumulators) so every W2 fragment is
// reused 4x (W2 is the 131 MB HBM stream that bounds the problem).
// A is staged to LDS in double-buffered 32 KB K-chunks by the Tensor Data
// Mover: wave 0 issues the DMA for chunk c+1, all waves compute on chunk c,
// wave 0 drains TENSORcnt, then the workgroup barrier publishes the buffer.
// The W2 fragment pipeline peels the chunk's last step so the loop body is
// branch-free; the cross-chunk preload is guarded once per chunk.
// ---------------------------------------------------------------------------
#define LCHUNK 128
#define NCHUNK (1024 / LCHUNK)

__global__ void __launch_bounds__(256)
wmma_logits(const float* __restrict__ A, const float* __restrict__ W2,
            const float* __restrict__ bias, float* __restrict__ C)
{
    __shared__ float ldsA[2][64 * LCHUNK];              // 2 x 32 KB

    const int lane = (int)(threadIdx.x & 31);
    const int wid  = (int)(threadIdx.x >> 5);           // 0..7
    const int half = lane >> 4;
    const int l16  = lane & 15;
    const int ncol = (int)blockIdx.x * 128 + wid * 16;  // this wave's column strip

    v8f acc0 = {}, acc1 = {}, acc2 = {}, acc3 = {};

    // W2 fragment pointer marches linearly across all of K.
    const float* bp = W2 + (size_t)ncol + l16 + (size_t)(2 * half) * 32000;
    v2f bv;
    bv.x = bp[0];
    bv.y = bp[32000];

#if HAVE_TDM
    if (wid == 0) {
        tdm_issue_tile_f32((unsigned int)(size_t)(uintptr_t)&ldsA[0][0],
                           (unsigned long long)(uintptr_t)A,
                           1024, 64, LCHUNK, 64, 1024);
        tdm_drain();
    }
    __syncthreads();
#endif

    for (int c = 0; c < NCHUNK; ++c) {
        const float* lbuf = &ldsA[c & 1][0];
#if HAVE_TDM
        if (wid == 0 && c + 1 < NCHUNK) {               // overlap DMA of c+1 with compute on c
            tdm_issue_tile_f32((unsigned int)(size_t)(uintptr_t)&ldsA[(c + 1) & 1][0],
                               (unsigned long long)(uintptr_t)(A + (c + 1) * LCHUNK),
                               1024, 64, LCHUNK, 64, 1024);
        }
#else
        __syncthreads();
        for (int idx = (int)threadIdx.x; idx < 64 * LCHUNK; idx += 256) {
            const int r = idx / LCHUNK, cc = idx - r * LCHUNK;
            ldsA[c & 1][idx] = A[r * 1024 + c * LCHUNK + cc];
        }
        __syncthreads();
#endif
        // Branch-free body: always preload next W2 fragment.
        for (int k = 0; k < LCHUNK - 4; k += 4) {
            if ((k & 31) == 0)                          // stream-ahead on W2
                __builtin_prefetch((const void*)(bp + (size_t)64 * 32000), 0, 0);
            const float* bpn = bp + (size_t)4 * 32000;
            v2f bn;
            bn.x = bpn[0];
            bn.y = bpn[32000];
            const int kc = k + 2 * half;
            const v2f a0 = *(const v2f*)&lbuf[( 0 + l16) * LCHUNK + kc];
            const v2f a1 = *(const v2f*)&lbuf[(16 + l16) * LCHUNK + kc];
            const v2f a2 = *(const v2f*)&lbuf[(32 + l16) * LCHUNK + kc];
            const v2f a3 = *(const v2f*)&lbuf[(48 + l16) * LCHUNK + kc];
            acc0 = __builtin_amdgcn_wmma_f32_16x16x4_f32(false, a0, false, bv, (short)0, acc0, false, false);
            acc1 = __builtin_amdgcn_wmma_f32_16x16x4_f32(false, a1, false, bv, (short)0, acc1, false, false);
            acc2 = __builtin_amdgcn_wmma_f32_16x16x4_f32(false, a2, false, bv, (short)0, acc2, false, false);
            acc3 = __builtin_amdgcn_wmma_f32_16x16x4_f32(false, a3, false, bv, (short)0, acc3, false, false);
            bv = bn;
            bp = bpn;
        }
        {   // Peeled last step of the chunk: cross-chunk preload guarded once.
            const float* bpn = bp + (size_t)4 * 32000;
            v2f bn = bv;
            if (c + 1 < NCHUNK) {
                bn.x = bpn[0];
                bn.y = bpn[32000];
            }
            const int kc = (LCHUNK - 4) + 2 * half;
            const v2f a0 = *(const v2f*)&lbuf[( 0 + l16) * LCHUNK + kc];
            const v2f a1 = *(const v2f*)&lbuf[(16 + l16) * LCHUNK + kc];
            const v2f a2 = *(const v2f*)&lbuf[(32 + l16) * LCHUNK + kc];
            const v2f a3 = *(const v2f*)&lbuf[(48 + l16) * LCHUNK + kc];
            acc0 = __builtin_amdgcn_wmma_f32_16x16x4_f32(false, a0, false, bv, (short)0, acc0, false, false);
            acc1 = __builtin_amdgcn_wmma_f32_16x16x4_f32(false, a1, false, bv, (short)0, acc1, false, false);
            acc2 = __builtin_amdgcn_wmma_f32_16x16x4_f32(false, a2, false, bv, (short)0, acc2, false, false);
            acc3 = __builtin_amdgcn_wmma_f32_16x16x4_f32(false, a3, false, bv, (short)0, acc3, false, false);
            bv = bn;
            bp = bpn;
        }
#if HAVE_TDM
        if (wid == 0) tdm_drain();                      // chunk c+1 landed
        __syncthreads();                                // publish; fences reuse of buf c&1
#endif
    }

    const float bval = bias[ncol + l16];
    float* cp = C + (size_t)ncol + l16;
    #pragma unroll
    for (int r = 0; r < 8; ++r) {
        const int m = r + 8 * half;
        cp[(size_t)(m +  0) * 32000] = acc0[r] + bval;
        cp[(size_t)(m + 16) * 32000] = acc1[r] + bval;
        cp[(size_t)(m + 32) * 32000] = acc2[r] + bval;
        cp[(size_t)(m + 48) * 32000] = acc3[r] + bval;
    }
}

// ---------------------------------------------------------------------------
// In-place softmax over rows of length `cols` (one wave32 per row).
// ---------------------------------------------------------------------------
__global__ void __launch_bounds__(256)
softmax_rows(float* __restrict__ a, int rows, int cols)
{
    const int wave = (int)((blockIdx.x * blockDim.x + threadIdx.x) >> 5);
    const int lane = (int)(threadIdx.x & 31);
    if (wave >= rows) return;
    float* row = a + (size_t)wave * cols;

    float mx = -INFINITY;
    for (int f = lane; f < cols; f += 32) mx = fmaxf(mx, row[f]);
    for (int o = 16; o > 0; o >>= 1) mx = fmaxf(mx, __shfl_xor(mx, o, 32));

    float s = 0.0f;
    for (int f = lane; f < cols; f += 32) {
        const float e = __expf(row[f] - mx);
        row[f] = e;
        s += e;
    }
    for (int o = 16; o > 0; o >>= 1) s += __shfl_xor(s, o, 32);
    const float inv = 1.0f / s;
    for (int f = lane; f < cols; f += 32) row[f] *= inv;
}

// ---------------------------------------------------------------------------
// context[i,f] = sum_b aw[i,b,f] * feat[i,b,f]; x = [context | emb[y]]
// ---------------------------------------------------------------------------
__global__ void __launch_bounds__(256)
context_embed(const float* __restrict__ aw, const float* __restrict__ feat,
              const int* __restrict__ y, const float* __restrict__ emb,
              float* __restrict__ x)
{
    const int idx = (int)(blockIdx.x * blockDim.x + threadIdx.x);   // 64*1024
    if (idx >= 64 * 1024) return;
    const int i = idx >> 10;
    const int j = idx & 1023;
    if (j < 512) {
        const float* ar = aw   + (size_t)i * 64 * 512 + j;
        const float* fr = feat + (size_t)i * 64 * 512 + j;
        float s = 0.0f;
        #pragma unroll 8
        for (int b = 0; b < 64; ++b) s += ar[b * 512] * fr[b * 512];
        x[idx] = s;
    } else {
        const int tok = y[i];
        x[idx] = emb[(size_t)tok * 512 + (j - 512)];
    }
}

// ---------------------------------------------------------------------------
// LSTM gate nonlinearity. z is (64 x 4096) = [zi | zf | zg | zo].
// ---------------------------------------------------------------------------
__global__ void __launch_bounds__(256)
lstm_act(const float* __restrict__ z, const float* __restrict__ c,
         float* __restrict__ h1, float* __restrict__ h2,
         float* __restrict__ cn, float* __restrict__ hws)
{
    const int idx = (int)(blockIdx.x * blockDim.x + threadIdx.x);   // 64*1024
    if (idx >= 64 * 1024) return;
    const int b = idx >> 10;
    const int u = idx & 1023;
    const float* zr = z + (size_t)b * 4096;
    const float zi = zr[u], zf = zr[u + 1024], zg = zr[u + 2048], zo = zr[u + 3072];
    const float ig = 1.0f / (1.0f + __expf(-zi));
    const float fg = 1.0f / (1.0f + __expf(-zf));
    const float gg = tanhf(zg);
    const float og = 1.0f / (1.0f + __expf(-zo));
    const float cnew = fg * c[idx] + ig * gg;
    const float hnew = og * tanhf(cnew);
    h1[idx] = hnew; h2[idx] = hnew; cn[idx] = cnew; hws[idx] = hnew;
}

// ---------------------------------------------------------------------------
// d_out (floats): logits[2048000] | h_new[65536] | h_new[65536]
//                 | c_new[65536] | aw[2097152]
// ---------------------------------------------------------------------------
extern "C" void kernel_launch(void* const* d_in, const int* in_sizes, int n_in,
                              void* d_out, int out_size, void* d_ws, size_t ws_size,
                              hipStream_t stream)
{
    (void)in_sizes; (void)n_in; (void)out_size; (void)ws_size;

    const int*   y    = (const int*)  d_in[0];
    const float* feat = (const float*)d_in[1];   // (64,64,512)
    const float* sout = (const float*)d_in[2];   // (64,1024)
    const float* h    = (const float*)d_in[3];   // (64,1024)
    const float* c    = (const float*)d_in[4];   // (64,1024)
    const float* Q    = (const float*)d_in[5];   // (1024,1024)
    const float* Km   = (const float*)d_in[6];   // (1024,64)
    const float* emb  = (const float*)d_in[7];   // (32000,512)
    const float* W_x  = (const float*)d_in[8];   // (1024,4096)
    const float* W_h  = (const float*)d_in[9];   // (1024,4096)
    const float* b_l  = (const float*)d_in[10];  // (4096,)
    const float* W1   = (const float*)d_in[11];  // (1024,1024)
    const float* b1   = (const float*)d_in[12];  // (1024,)
    const float* W2   = (const float*)d_in[13];  // (1024,32000)
    const float* b2   = (const float*)d_in[14];  // (32000,)

    float* out    = (float*)d_out;
    float* logits = out;                           // 64*32000
    float* hn1    = out + 2048000;
    float* hn2    = out + 2048000 + 65536;
    float* cn     = out + 2048000 + 2 * 65536;
    float* aw     = out + 2048000 + 3 * 65536;     // 64*64*512 (softmax in place)

    float* ws   = (float*)d_ws;
    float* tmp1 = ws;                              // 64*1024
    float* wbuf = ws + 65536;                      // 64*64
    float* xbuf = ws + 69632;                      // 64*1024 (x, later h_new)
    float* zbuf = ws + 135168;                     // 64*4096
    float* tbuf = ws + 397312;                     // 64*1024

    const dim3 blk(256);

    // 1) tmp1 = state_output @ Q^T   (B[k][n] = Q[n][k] -> bk=1, bn=1024)
    wmma_gemm64<<<32, blk, 0, stream>>>(sout, 1024, 0, Q, 1, 1024, 0,
                                        nullptr, nullptr, nullptr,
                                        tmp1, 1024, 0, 1024, 1024, 1);
    // 2) w = tmp1 @ K
    wmma_gemm64<<<2, blk, 0, stream>>>(tmp1, 1024, 0, Km, 64, 1, 0,
                                       nullptr, nullptr, nullptr,
                                       wbuf, 64, 0, 64, 1024, 1);
    // 3) aw_logits[i] = w @ features[i]  (batched over i, into d_out slice)
    wmma_gemm64<<<1024, blk, 0, stream>>>(wbuf, 64, 0, feat, 512, 1, 32768,
                                          nullptr, nullptr, nullptr,
                                          aw, 512, 32768, 512, 64, 64);
    // 4) softmax over f (4096 rows of 512)
    softmax_rows<<<512, blk, 0, stream>>>(aw, 4096, 512);
    // 5) context + embedding gather -> x
    context_embed<<<256, blk, 0, stream>>>(aw, feat, y, emb, xbuf);
    // 6) z = x@W_x + h@W_h + b_lstm
    wmma_gemm64<<<128, blk, 0, stream>>>(xbuf, 1024, 0, W_x, 4096, 1, 0,
                                         h, W_h, b_l,
                                         zbuf, 4096, 0, 4096, 1024, 1);
    // 7) gates -> h_new (two output copies + ws), c_new
    lstm_act<<<256, blk, 0, stream>>>(zbuf, c, hn1, hn2, cn, xbuf);
    // 8) t = h_new @ W1 + b1
    wmma_gemm64<<<32, blk, 0, stream>>>(xbuf, 1024, 0, W1, 1024, 1, 0,
                                        nullptr, nullptr, b1,
                                        tbuf, 1024, 0, 1024, 1024, 1);
    // 9) logits = t @ W2 + b2  (double-buffered TDM staging, 4x W2 reuse)
    wmma_logits<<<250, blk, 0, stream>>>(tbuf, W2, b2, logits);
}